// PointNetPlusPlusCls_55456617726263
// MI455X (gfx1250) — compile-verified
//
#include <hip/hip_runtime.h>
#include <hip/hip_bf16.h>
#include <math.h>

// ---------------------------------------------------------------------------
// PointNet++ classification forward for MI455X (gfx1250).
// All GEMMs use V_WMMA_F32_16X16X4_F32 (exact fp32 semantics vs reference).
// sa1/sa2 are fully fused per-group kernels (group build + 3 MLP layers +
// BN/ReLU + maxpool all in LDS). sa3/FC use a generic 8-wave WMMA GEMM.
// FPS is a per-batch LDS-resident serial scan (the latency bottleneck).
// ---------------------------------------------------------------------------

#define BB      16
#define NN      4096
#define NP1     512
#define NS1     32
#define NP2     128
#define NS2     64
#define NCLS    40
#define BN_EPS  1e-5f

typedef float v2f __attribute__((ext_vector_type(2)));
typedef float v8f __attribute__((ext_vector_type(8)));

struct BNLayer {
    const float* W;   // (K, N) row-major
    const float* b;
    const float* g;   // gamma
    const float* be;  // beta
    const float* mu;  // mean
    const float* var; // var
};

// ---- 16x16 output tile: A(MxK, lda) @ W(KxN, ldw), K stepped by 4 ----------
// A frag layout (ISA 7.12.2, 32-bit A 16x4): lanes 0-15 hold K=k0,k0+1;
// lanes 16-31 hold K=k0+2,k0+3.  B mirrored (row-striped across lanes).
// Main K loop is unguarded (no exec juggling); K%4 handled in a single tail
// step.  Out-of-range columns are CLAMPED (not branched) -- garbage lanes are
// discarded by the store-side guard, keeping EXEC all-ones around the WMMA.
__device__ __forceinline__ v8f wmma_tile_f32(const float* __restrict__ A, int lda,
                                             const float* __restrict__ W, int ldw,
                                             int K, int m0, int n0, int lane, int Ncols)
{
    v8f acc = {0.f, 0.f, 0.f, 0.f, 0.f, 0.f, 0.f, 0.f};
    const int half2 = (lane >> 4) << 1;  // 0 or 2
    const int l16   = lane & 15;
    const int col   = n0 + l16;
    const int colc  = (col < Ncols) ? col : (Ncols - 1);  // clamp, no divergence
    const float* __restrict__ Arow = A + (size_t)(m0 + l16) * lda;

    const int Kmain = K & ~3;
    int k0 = 0;
    for (; k0 < Kmain; k0 += 4) {
        const int ka = k0 + half2;
        v2f a, b;
        a.x = Arow[ka];
        a.y = Arow[ka + 1];
        b.x = W[(size_t)(ka    ) * ldw + colc];
        b.y = W[(size_t)(ka + 1) * ldw + colc];
        acc = __builtin_amdgcn_wmma_f32_16x16x4_f32(false, a, false, b,
                                                    (short)0, acc, false, false);
    }
    if (k0 < K) {  // tail: K % 4 in {1,2,3}
        const int ka = k0 + half2;
        v2f a, b;
        a.x = (ka     < K) ? Arow[ka    ] : 0.0f;
        a.y = (ka + 1 < K) ? Arow[ka + 1] : 0.0f;
        b.x = (ka     < K) ? W[(size_t)(ka    ) * ldw + colc] : 0.0f;
        b.y = (ka + 1 < K) ? W[(size_t)(ka + 1) * ldw + colc] : 0.0f;
        acc = __builtin_amdgcn_wmma_f32_16x16x4_f32(false, a, false, b,
                                                    (short)0, acc, false, false);
    }
    return acc;
}

// ---- epilogue: bias + optional BN + optional ReLU, store 16x16 tile --------
// C/D layout: VGPR j -> row m0+j (lanes 0-15) / m0+j+8 (lanes 16-31).
__device__ __forceinline__ void store_tile(v8f acc, float* __restrict__ Out, int ldo,
                                           int m0, int n0, int lane,
                                           const BNLayer& L, int doBN, int doRelu, int Ncols)
{
    const int half = lane >> 4;
    const int col  = n0 + (lane & 15);
    if (col < Ncols) {
        const float bb = L.b ? L.b[col] : 0.0f;
        float g = 1.f, be = 0.f, mu = 0.f, iv = 1.f;
        if (doBN) {
            g  = L.g[col];  be = L.be[col];
            mu = L.mu[col]; iv = rsqrtf(L.var[col] + BN_EPS);
        }
#pragma unroll
        for (int j = 0; j < 8; ++j) {
            const int row = m0 + j + half * 8;
            float v = acc[j] + bb;
            if (doBN)   v = g * (v - mu) * iv + be;
            if (doRelu) v = fmaxf(v, 0.0f);
            Out[(size_t)row * ldo + col] = v;
        }
    }
}

// ---------------------------------------------------------------------------
// x (B,6,N) -> xyz (B,N,3), pts (B,N,3)
__global__ void k_split(const float* __restrict__ x, float* __restrict__ xyz,
                        float* __restrict__ pts)
{
    int g = blockIdx.x * blockDim.x + threadIdx.x;
    if (g >= BB * NN) return;
    int b = g / NN, n = g % NN;
    const float* xb = x + (size_t)b * 6 * NN;
#pragma unroll
    for (int c = 0; c < 3; ++c) {
        xyz[((size_t)b * NN + n) * 3 + c] = xb[(size_t)c * NN + n];
        pts[((size_t)b * NN + n) * 3 + c] = xb[(size_t)(c + 3) * NN + n];
    }
}

// ---------------------------------------------------------------------------
// Farthest point sampling: one block per batch, LDS-resident distances.
// Emits new_xyz directly (fps indices are only used for the gather).
__global__ void k_fps(const float* __restrict__ xyz, int N, int npoint,
                      float* __restrict__ new_xyz)
{
    __shared__ float dist[4096];
    __shared__ float rv[256];
    __shared__ int   ri[256];
    __shared__ int   s_far;

    const int b   = blockIdx.x;
    const int tid = threadIdx.x;
    const float* xb = xyz + (size_t)b * N * 3;

    for (int n = tid; n < N; n += 256) dist[n] = 1e10f;
    if (tid == 0) s_far = 0;
    __syncthreads();

    for (int s = 0; s < npoint; ++s) {
        const int far = s_far;
        if (tid == 0) {
            new_xyz[((size_t)b * npoint + s) * 3 + 0] = xb[far * 3 + 0];
            new_xyz[((size_t)b * npoint + s) * 3 + 1] = xb[far * 3 + 1];
            new_xyz[((size_t)b * npoint + s) * 3 + 2] = xb[far * 3 + 2];
        }
        const float cx = xb[far * 3 + 0];
        const float cy = xb[far * 3 + 1];
        const float cz = xb[far * 3 + 2];

        float lv = -1.0f; int li = 0;
        for (int n = tid; n < N; n += 256) {
            float dx = xb[n * 3 + 0] - cx;
            float dy = xb[n * 3 + 1] - cy;
            float dz = xb[n * 3 + 2] - cz;
            float d  = dx * dx + dy * dy + dz * dz;
            float nd = fminf(dist[n], d);
            dist[n]  = nd;
            if (nd > lv) { lv = nd; li = n; }
        }
        rv[tid] = lv; ri[tid] = li;
        __syncthreads();
        for (int st = 128; st > 0; st >>= 1) {
            if (tid < st) {
                float ov = rv[tid + st]; int oi = ri[tid + st];
                if (ov > rv[tid] || (ov == rv[tid] && oi < ri[tid])) {
                    rv[tid] = ov; ri[tid] = oi;
                }
            }
            __syncthreads();
        }
        if (tid == 0) s_far = ri[0];
        __syncthreads();
    }
}

// ---------------------------------------------------------------------------
// Ball query: one thread per centroid; scan in index order and take the first
// nsample points within radius (== reference sort-and-take + fill-with-first).
__global__ void k_query_ball(const float* __restrict__ xyz,
                             const float* __restrict__ new_xyz,
                             int N, int npoint, int nsample, float r2,
                             int* __restrict__ gi)
{
    const int b = blockIdx.x;
    const int s = threadIdx.x;
    if (s >= npoint) return;
    const float* xb = xyz + (size_t)b * N * 3;
    const float cx = new_xyz[((size_t)b * npoint + s) * 3 + 0];
    const float cy = new_xyz[((size_t)b * npoint + s) * 3 + 1];
    const float cz = new_xyz[((size_t)b * npoint + s) * 3 + 2];
    int* row = gi + ((size_t)b * npoint + s) * nsample;
    int cnt = 0, first = 0;
    for (int n = 0; n < N && cnt < nsample; ++n) {
        float dx = xb[n * 3 + 0] - cx;
        float dy = xb[n * 3 + 1] - cy;
        float dz = xb[n * 3 + 2] - cz;
        if (dx * dx + dy * dy + dz * dz <= r2) {
            if (cnt == 0) first = n;
            row[cnt++] = n;
        }
    }
    for (int j = cnt; j < nsample; ++j) row[j] = first;
}

// ---------------------------------------------------------------------------
// Fused SA1: one block per (b, centroid). 32 samples, cin=6 -> 64 -> 64 -> 128,
// BN+ReLU each layer, maxpool over the 32 samples. All intermediates in LDS.
__global__ void k_sa1(const float* __restrict__ xyz, const float* __restrict__ pts,
                      const float* __restrict__ nxyz, const int* __restrict__ gi,
                      BNLayer L0, BNLayer L1, BNLayer L2,
                      float* __restrict__ out)
{
    __shared__ float X[32 * 64];   // input (ld 8) then layer-1 output (ld 64)
    __shared__ float Y[32 * 128];  // layer-0 output (ld 64) then layer-2 (ld 128)

    const int grp = blockIdx.x;
    const int b   = grp / NP1;
    const int tid = threadIdx.x;
    const int wave = tid >> 5, lane = tid & 31;
    const int* gp = gi + (size_t)grp * NS1;
    const float cx = nxyz[(size_t)grp * 3 + 0];
    const float cy = nxyz[(size_t)grp * 3 + 1];
    const float cz = nxyz[(size_t)grp * 3 + 2];

    // Build grouped features: [xyz - center, pts], padded to stride 8.
    for (int idx = tid; idx < 32 * 8; idx += 256) {
        const int j = idx >> 3, c = idx & 7;
        const int n = gp[j];
        float v = 0.0f;
        if (c < 3)       v = xyz[((size_t)b * NN + n) * 3 + c] - (c == 0 ? cx : (c == 1 ? cy : cz));
        else if (c < 6)  v = pts[((size_t)b * NN + n) * 3 + (c - 3)];
        X[j * 8 + c] = v;
    }
    __syncthreads();

    // L0: X(32x6, ld8) @ W0(6x64) -> Y(32x64).  8 tiles, one per wave.
    { int tm = wave & 1, tn = wave >> 1;
      v8f a = wmma_tile_f32(X, 8, L0.W, 64, 6, tm * 16, tn * 16, lane, 64);
      store_tile(a, Y, 64, tm * 16, tn * 16, lane, L0, 1, 1, 64); }
    __syncthreads();

    // L1: Y(32x64) @ W1(64x64) -> X(32x64)
    { int tm = wave & 1, tn = wave >> 1;
      v8f a = wmma_tile_f32(Y, 64, L1.W, 64, 64, tm * 16, tn * 16, lane, 64);
      store_tile(a, X, 64, tm * 16, tn * 16, lane, L1, 1, 1, 64); }
    __syncthreads();

    // L2: X(32x64) @ W2(64x128) -> Y(32x128).  16 tiles, 2 per wave.
    for (int t = wave; t < 16; t += 8) {
        int tm = t & 1, tn = t >> 1;
        v8f a = wmma_tile_f32(X, 64, L2.W, 128, 64, tm * 16, tn * 16, lane, 128);
        store_tile(a, Y, 128, tm * 16, tn * 16, lane, L2, 1, 1, 128);
    }
    __syncthreads();

    // Maxpool over 32 samples.
    for (int c = tid; c < 128; c += 256) {
        float m = Y[c];
        for (int j = 1; j < 32; ++j) m = fmaxf(m, Y[j * 128 + c]);
        out[(size_t)grp * 128 + c] = m;
    }
}

// ---------------------------------------------------------------------------
// Fused SA2: one block per (b, centroid). 64 samples, cin=131 -> 128 -> 128 -> 256.
__global__ void k_sa2(const float* __restrict__ xyz,   // l1_xyz (B,512,3)
                      const float* __restrict__ feat,  // l1_pts (B,512,128)
                      const float* __restrict__ nxyz,  // (B,128,3)
                      const int* __restrict__ gi,
                      BNLayer L0, BNLayer L1, BNLayer L2,
                      float* __restrict__ out)
{
    __shared__ float X[64 * 132];  // input (ld 132) then layer-1 output (ld 128)
    __shared__ float Y[64 * 256];  // layer-0 output (ld 128) then layer-2 (ld 256)

    const int grp = blockIdx.x;
    const int b   = grp / NP2;
    const int tid = threadIdx.x;
    const int wave = tid >> 5, lane = tid & 31;
    const int* gp = gi + (size_t)grp * NS2;
    const float cx = nxyz[(size_t)grp * 3 + 0];
    const float cy = nxyz[(size_t)grp * 3 + 1];
    const float cz = nxyz[(size_t)grp * 3 + 2];

    for (int idx = tid; idx < 64 * 132; idx += 256) {
        const int j = idx / 132, c = idx % 132;
        const int n = gp[j];
        float v = 0.0f;
        if (c < 3)        v = xyz[((size_t)b * NP1 + n) * 3 + c] - (c == 0 ? cx : (c == 1 ? cy : cz));
        else if (c < 131) v = feat[((size_t)b * NP1 + n) * 128 + (c - 3)];
        X[j * 132 + c] = v;
    }
    __syncthreads();

    // L0: X(64x131, ld132) @ W0(131x128) -> Y(64x128).  32 tiles, 4 per wave.
    for (int t = wave; t < 32; t += 8) {
        int tm = t & 3, tn = t >> 2;
        v8f a = wmma_tile_f32(X, 132, L0.W, 128, 131, tm * 16, tn * 16, lane, 128);
        store_tile(a, Y, 128, tm * 16, tn * 16, lane, L0, 1, 1, 128);
    }
    __syncthreads();

    // L1: Y(64x128) @ W1(128x128) -> X(64x128)
    for (int t = wave; t < 32; t += 8) {
        int tm = t & 3, tn = t >> 2;
        v8f a = wmma_tile_f32(Y, 128, L1.W, 128, 128, tm * 16, tn * 16, lane, 128);
        store_tile(a, X, 128, tm * 16, tn * 16, lane, L1, 1, 1, 128);
    }
    __syncthreads();

    // L2: X(64x128) @ W2(128x256) -> Y(64x256).  64 tiles, 8 per wave.
    for (int t = wave; t < 64; t += 8) {
        int tm = t & 3, tn = t >> 2;
        v8f a = wmma_tile_f32(X, 128, L2.W, 256, 128, tm * 16, tn * 16, lane, 256);
        store_tile(a, Y, 256, tm * 16, tn * 16, lane, L2, 1, 1, 256);
    }
    __syncthreads();

    for (int c = tid; c < 256; c += 256) {
        float m = Y[c];
        for (int j = 1; j < 64; ++j) m = fmaxf(m, Y[j * 256 + c]);
        out[(size_t)grp * 256 + c] = m;
    }
}

// ---------------------------------------------------------------------------
// Generic WMMA GEMM + fused bias/BN/ReLU.  8 waves/block, each wave owns one
// of 8 adjacent 16-wide N tiles (uniform per-wave branch keeps EXEC all-ones).
__global__ void __launch_bounds__(256)
k_gemm(const float* __restrict__ A, BNLayer L, float* __restrict__ Out,
       int M, int N, int K, int doBN, int doRelu)
{
    const int wave = threadIdx.x >> 5, lane = threadIdx.x & 31;
    const int tm = blockIdx.y;
    const int tn = blockIdx.x * 8 + wave;
    if (tn * 16 < N) {
        v8f acc = wmma_tile_f32(A, K, L.W, N, K, tm * 16, tn * 16, lane, N);
        store_tile(acc, Out, N, tm * 16, tn * 16, lane, L, doBN, doRelu, N);
    }
}

// ---------------------------------------------------------------------------
// in3 row (b*128+j): [l2_xyz (3) | l2_pts (256)]
__global__ void k_concat3(const float* __restrict__ nxyz2, const float* __restrict__ l2pts,
                          float* __restrict__ in3)
{
    int g = blockIdx.x * blockDim.x + threadIdx.x;
    const int total = BB * NP2 * 259;
    if (g >= total) return;
    int row = g / 259, c = g % 259;
    in3[g] = (c < 3) ? nxyz2[(size_t)row * 3 + c] : l2pts[(size_t)row * 256 + (c - 3)];
}

// (B,128,1024) -> (B,1024) max over points
__global__ void k_pool_global(const float* __restrict__ t2, float* __restrict__ g)
{
    const int b = blockIdx.x;
    for (int c = threadIdx.x; c < 1024; c += 256) {
        float m = t2[((size_t)b * NP2) * 1024 + c];
        for (int j = 1; j < NP2; ++j)
            m = fmaxf(m, t2[((size_t)b * NP2 + j) * 1024 + c]);
        g[(size_t)b * 1024 + c] = m;
    }
}

__global__ void k_logsoftmax(const float* __restrict__ logits, float* __restrict__ out)
{
    const int r = threadIdx.x;
    if (r < BB) {
        float mx = -1e30f;
        for (int c = 0; c < NCLS; ++c) mx = fmaxf(mx, logits[r * NCLS + c]);
        float s = 0.0f;
        for (int c = 0; c < NCLS; ++c) s += expf(logits[r * NCLS + c] - mx);
        const float ls = logf(s);
        for (int c = 0; c < NCLS; ++c)
            out[r * NCLS + c] = logits[r * NCLS + c] - mx - ls;
    }
}

// ---------------------------------------------------------------------------
static inline size_t alignUp(size_t v, size_t a) { return (v + a - 1) / a * a; }

extern "C" void kernel_launch(void* const* d_in, const int* in_sizes, int n_in,
                              void* d_out, int out_size, void* d_ws, size_t ws_size,
                              hipStream_t stream)
{
    (void)in_sizes; (void)n_in; (void)out_size; (void)ws_size;

    // Input order: setup_inputs() insertion order, nested dicts flattened in
    // insertion order:
    //   0: x
    //   sa1 layers 0..2 at 1+6j : {W,b,gamma,beta,mean,var}
    //   sa2 at 19+6j, sa3 at 37+6j, fc1 at 55, fc2 at 61, fc3 {W,b} at 67,68
    auto P = [&](int i) { return (const float*)d_in[i]; };
    auto layer = [&](int base) {
        BNLayer L; L.W = P(base); L.b = P(base + 1); L.g = P(base + 2);
        L.be = P(base + 3); L.mu = P(base + 4); L.var = P(base + 5); return L;
    };
    BNLayer sa1L0 = layer(1),  sa1L1 = layer(7),  sa1L2 = layer(13);
    BNLayer sa2L0 = layer(19), sa2L1 = layer(25), sa2L2 = layer(31);
    BNLayer sa3L0 = layer(37), sa3L1 = layer(43), sa3L2 = layer(49);
    BNLayer fc1   = layer(55), fc2   = layer(61);
    BNLayer fc3;  fc3.W = P(67); fc3.b = P(68);
    fc3.g = fc3.be = fc3.mu = fc3.var = nullptr;

    // Workspace carve (floats/ints), 256B aligned.
    char* w = (char*)d_ws; size_t off = 0;
    auto cF = [&](size_t n) { float* p = (float*)(w + off); off += alignUp(n * 4, 256); return p; };
    auto cI = [&](size_t n) { int*   p = (int*)  (w + off); off += alignUp(n * 4, 256); return p; };

    float* xyz    = cF((size_t)BB * NN * 3);
    float* pts    = cF((size_t)BB * NN * 3);
    float* nxyz1  = cF((size_t)BB * NP1 * 3);
    int*   gi1    = cI((size_t)BB * NP1 * NS1);
    float* l1pts  = cF((size_t)BB * NP1 * 128);
    float* nxyz2  = cF((size_t)BB * NP2 * 3);
    int*   gi2    = cI((size_t)BB * NP2 * NS2);
    float* l2pts  = cF((size_t)BB * NP2 * 256);
    float* in3    = cF((size_t)BB * NP2 * 259);
    float* t0     = cF((size_t)BB * NP2 * 256);
    float* t1     = cF((size_t)BB * NP2 * 512);
    float* t2     = cF((size_t)BB * NP2 * 1024);
    float* gfeat  = cF((size_t)BB * 1024);
    float* f1     = cF((size_t)BB * 512);
    float* f2     = cF((size_t)BB * 256);
    float* logits = cF((size_t)BB * NCLS);

    const float* x = P(0);
    float* out = (float*)d_out;

    // 1. split input
    k_split<<<(BB * NN + 255) / 256, 256, 0, stream>>>(x, xyz, pts);

    // 2. SA1: FPS -> ball query -> fused group-MLP-pool
    k_fps<<<BB, 256, 0, stream>>>(xyz, NN, NP1, nxyz1);
    k_query_ball<<<BB, NP1, 0, stream>>>(xyz, nxyz1, NN, NP1, NS1, 0.2f * 0.2f, gi1);
    k_sa1<<<BB * NP1, 256, 0, stream>>>(xyz, pts, nxyz1, gi1, sa1L0, sa1L1, sa1L2, l1pts);

    // 3. SA2
    k_fps<<<BB, 256, 0, stream>>>(nxyz1, NP1, NP2, nxyz2);
    k_query_ball<<<BB, NP2, 0, stream>>>(nxyz1, nxyz2, NP1, NP2, NS2, 0.4f * 0.4f, gi2);
    k_sa2<<<BB * NP2, 256, 0, stream>>>(nxyz1, l1pts, nxyz2, gi2, sa2L0, sa2L1, sa2L2, l2pts);

    // 4. SA3 (group all): concat -> 3x GEMM+BN+ReLU -> maxpool
    k_concat3<<<(BB * NP2 * 259 + 255) / 256, 256, 0, stream>>>(nxyz2, l2pts, in3);
    {   const int M = BB * NP2;
        k_gemm<<<dim3(256 / 128,  M / 16), 256, 0, stream>>>(in3, sa3L0, t0, M, 256,  259, 1, 1);
        k_gemm<<<dim3(512 / 128,  M / 16), 256, 0, stream>>>(t0,  sa3L1, t1, M, 512,  256, 1, 1);
        k_gemm<<<dim3(1024 / 128, M / 16), 256, 0, stream>>>(t1,  sa3L2, t2, M, 1024, 512, 1, 1);
    }
    k_pool_global<<<BB, 256, 0, stream>>>(t2, gfeat);

    // 5. FC head
    k_gemm<<<dim3(4, 1), 256, 0, stream>>>(gfeat, fc1, f1, BB, 512, 1024, 1, 1);
    k_gemm<<<dim3(2, 1), 256, 0, stream>>>(f1,    fc2, f2, BB, 256, 512,  1, 1);
    k_gemm<<<dim3(1, 1), 256, 0, stream>>>(f2,    fc3, logits, BB, NCLS, 256, 0, 0);

    // 6. log_softmax
    k_logsoftmax<<<1, 32, 0, stream>>>(logits, out);
}